// ChebNet_16123307229541
// MI455X (gfx1250) — compile-verified
//
#include <hip/hip_runtime.h>
#include <hip/hip_bf16.h>

typedef __attribute__((ext_vector_type(2))) float v2f;
typedef __attribute__((ext_vector_type(8))) float v8f;

#define NFEAT   256
#define NHIDDEN 128
#define NCLASS  40
#define NCPAD   48

// ---------------------------------------------------------------------------
// FC1: h = relu(x @ W1 + b1), poly = theta0 * h.
// One wave computes one 16x16 output tile via V_WMMA_F32_16X16X4_F32.
// Block = 256 threads = 8 waves = the 8 column tiles of NHIDDEN=128.
// Grid = N/16 row tiles.  N % 16 == 0 (100000 = 6250*16), no masking needed.
// ---------------------------------------------------------------------------
__global__ void fc1_relu_poly0_kernel(const float* __restrict__ x,
                                      const float* __restrict__ W1,
                                      const float* __restrict__ b1,
                                      const float* __restrict__ thetas,
                                      float* __restrict__ h,
                                      float* __restrict__ poly) {
    const int lane = threadIdx.x & 31;
    const int wv   = threadIdx.x >> 5;     // column tile 0..7
    const int hf   = lane >> 4;            // half-wave: 0 or 1
    const int m    = lane & 15;            // A row within tile / B,C,D col lane
    const int row0 = blockIdx.x * 16;
    const int n    = wv * 16 + m;          // output column for this lane

    v8f c = {};
    for (int k = 0; k < NFEAT; k += 4) {
        // A 16x4 f32: VGPR0 = {K=0 | K=2}, VGPR1 = {K=1 | K=3}
        const float* ap = &x[(size_t)(row0 + m) * NFEAT + k + 2 * hf];
        v2f a; a.x = ap[0]; a.y = ap[1];
        // B 4x16 f32: VGPR0 = {K=0 | K=2}, VGPR1 = {K=1 | K=3}
        const float* bp = &W1[(size_t)(k + 2 * hf) * NHIDDEN + n];
        v2f b; b.x = bp[0]; b.y = bp[NHIDDEN];
        c = __builtin_amdgcn_wmma_f32_16x16x4_f32(false, a, false, b,
                                                  (short)0, c, false, false);
    }
    const float bias = b1[n];
    const float th0  = thetas[0];
    for (int v = 0; v < 8; ++v) {
        float val = c[v] + bias;
        val = val > 0.0f ? val : 0.0f;
        const size_t idx = (size_t)(row0 + v + 8 * hf) * NHIDDEN + n;
        h[idx]    = val;
        poly[idx] = th0 * val;
    }
}

// ---------------------------------------------------------------------------
// SpMM scatter: out[row[e]] += scale * w[e] * t[col[e]]  over 128 features.
// One wave per edge (grid-stride); each lane handles 4 features (float4).
// t (51.2 MB) fits in the 192 MB L2, so the random gathers are L2-resident;
// float atomics resolve in the L2 atomic units.
// ---------------------------------------------------------------------------
__global__ void spmm_atomic_kernel(const int*   __restrict__ erow,
                                   const int*   __restrict__ ecol,
                                   const float* __restrict__ ew,
                                   const float* __restrict__ t,
                                   float*       __restrict__ out,
                                   float scale, int E) {
    const int lane   = threadIdx.x & 31;
    const int wavesPerBlk = blockDim.x >> 5;
    const int wgid   = blockIdx.x * wavesPerBlk + (threadIdx.x >> 5);
    const int nwaves = gridDim.x * wavesPerBlk;
    const int f      = lane * 4;

    for (int e = wgid; e < E; e += nwaves) {
        const int   r  = erow[e];
        const int   cc = ecol[e];
        const float w  = ew[e] * scale;
        const float4 tv = *(const float4*)&t[(size_t)cc * NHIDDEN + f];
        float* o = &out[(size_t)r * NHIDDEN + f];
        unsafeAtomicAdd(o + 0, w * tv.x);
        unsafeAtomicAdd(o + 1, w * tv.y);
        unsafeAtomicAdd(o + 2, w * tv.z);
        unsafeAtomicAdd(o + 3, w * tv.w);
    }
}

// ---------------------------------------------------------------------------
// Elementwise helpers.
// ---------------------------------------------------------------------------
__global__ void negate_inplace_kernel(float* __restrict__ a, size_t n) {
    size_t i = (size_t)blockIdx.x * blockDim.x + threadIdx.x;
    if (i < n) a[i] = -a[i];
}

__global__ void axpy_theta_kernel(float* __restrict__ poly,
                                  const float* __restrict__ t,
                                  const float* __restrict__ thetas,
                                  int ti, size_t n) {
    size_t i = (size_t)blockIdx.x * blockDim.x + threadIdx.x;
    if (i < n) poly[i] += thetas[ti] * t[i];
}

// ---------------------------------------------------------------------------
// FC2: ypad[N,48] = poly @ W2 + b2 (cols >= 40 garbage/zero-bias).
// One wave per 16-row tile, holding 3 column-tile accumulators (cols 0..47).
// Block = 64 threads = 2 waves; grid = 6250/2 = 3125.
// ---------------------------------------------------------------------------
__global__ void fc2_wmma_kernel(const float* __restrict__ poly,
                                const float* __restrict__ W2,
                                const float* __restrict__ b2,
                                float* __restrict__ ypad) {
    const int lane = threadIdx.x & 31;
    const int wv   = threadIdx.x >> 5;     // 0 or 1
    const int hf   = lane >> 4;
    const int m    = lane & 15;
    const int row0 = (blockIdx.x * 2 + wv) * 16;

    const int  n0 = 0  + m, n1 = 16 + m, n2 = 32 + m;
    const bool v2ok = (n2 < NCLASS);       // mask cols 40..47

    v8f c0 = {}, c1 = {}, c2 = {};
    for (int k = 0; k < NHIDDEN; k += 4) {
        const float* ap = &poly[(size_t)(row0 + m) * NHIDDEN + k + 2 * hf];
        v2f a; a.x = ap[0]; a.y = ap[1];

        const int kr0 = (k + 2 * hf) * NCLASS;
        const int kr1 = kr0 + NCLASS;
        v2f b0, b1v, b2v;
        b0.x  = W2[kr0 + n0];              b0.y  = W2[kr1 + n0];
        b1v.x = W2[kr0 + n1];              b1v.y = W2[kr1 + n1];
        b2v.x = v2ok ? W2[kr0 + n2] : 0.f; b2v.y = v2ok ? W2[kr1 + n2] : 0.f;

        c0 = __builtin_amdgcn_wmma_f32_16x16x4_f32(false, a, false, b0,
                                                   (short)0, c0, false, false);
        c1 = __builtin_amdgcn_wmma_f32_16x16x4_f32(false, a, false, b1v,
                                                   (short)0, c1, false, false);
        c2 = __builtin_amdgcn_wmma_f32_16x16x4_f32(false, a, false, b2v,
                                                   (short)0, c2, false, false);
    }
    const float bb0 = b2[n0];
    const float bb1 = b2[n1];
    const float bb2 = v2ok ? b2[n2] : 0.0f;
    for (int v = 0; v < 8; ++v) {
        const size_t rbase = (size_t)(row0 + v + 8 * hf) * NCPAD;
        ypad[rbase + n0] = c0[v] + bb0;
        ypad[rbase + n1] = c1[v] + bb1;
        ypad[rbase + n2] = c2[v] + bb2;
    }
}

// ---------------------------------------------------------------------------
// Row-wise log-softmax over 40 classes. One thread per row (tiny traffic).
// ---------------------------------------------------------------------------
__global__ void log_softmax40_kernel(const float* __restrict__ ypad,
                                     float* __restrict__ out, int N) {
    const int r = blockIdx.x * blockDim.x + threadIdx.x;
    if (r >= N) return;
    const float* y = &ypad[(size_t)r * NCPAD];
    float mx = -3.402823466e38f;
    for (int j = 0; j < NCLASS; ++j) mx = fmaxf(mx, y[j]);
    float s = 0.0f;
    for (int j = 0; j < NCLASS; ++j) s += expf(y[j] - mx);
    const float ls = logf(s);
    float* o = &out[(size_t)r * NCLASS];
    for (int j = 0; j < NCLASS; ++j) o[j] = y[j] - mx - ls;
}

// ---------------------------------------------------------------------------
// Driver.
// ---------------------------------------------------------------------------
extern "C" void kernel_launch(void* const* d_in, const int* in_sizes, int n_in,
                              void* d_out, int out_size, void* d_ws, size_t ws_size,
                              hipStream_t stream) {
    const float* x      = (const float*)d_in[0];
    const int*   erow   = (const int*)  d_in[1];
    const int*   ecol   = (const int*)  d_in[2];
    const float* ew     = (const float*)d_in[3];
    const float* W1     = (const float*)d_in[4];
    const float* b1     = (const float*)d_in[5];
    const float* W2     = (const float*)d_in[6];
    const float* b2     = (const float*)d_in[7];
    const float* thetas = (const float*)d_in[8];

    const int N = in_sizes[0] / NFEAT;     // 100000
    const int E = in_sizes[1];             // 1600000
    const size_t nh = (size_t)N * NHIDDEN; // 12.8M floats

    float* bufA = (float*)d_ws;            // t0, then t2, t4, t6
    float* bufB = bufA + nh;               // t1, then t3, t5, t7
    float* poly = bufB + nh;
    float* ypad = poly + nh;               // [N, 48]

    const int rowTiles = N / 16;           // 6250
    const int ewBlocks = 4096;
    const int ewGrid   = (int)((nh + 255) / 256);

    // 1) h = relu(x@W1+b1) -> bufA ; poly = theta0 * h
    fc1_relu_poly0_kernel<<<rowTiles, 256, 0, stream>>>(x, W1, b1, thetas, bufA, poly);

    // 2) t1 = L @ t0 -> bufB ; poly += theta1 * t1
    hipMemsetAsync(bufB, 0, nh * sizeof(float), stream);
    spmm_atomic_kernel<<<ewBlocks, 256, 0, stream>>>(erow, ecol, ew, bufA, bufB, 1.0f, E);
    axpy_theta_kernel<<<ewGrid, 256, 0, stream>>>(poly, bufB, thetas, 1, nh);

    // 3) t_i = 2*L@prev0 - prev1, fused: negate prev1 in place, atomic-add 2*L@prev0
    float* prev0 = bufB;
    float* prev1 = bufA;
    for (int i = 2; i < 8; ++i) {
        negate_inplace_kernel<<<ewGrid, 256, 0, stream>>>(prev1, nh);
        spmm_atomic_kernel<<<ewBlocks, 256, 0, stream>>>(erow, ecol, ew, prev0, prev1, 2.0f, E);
        axpy_theta_kernel<<<ewGrid, 256, 0, stream>>>(poly, prev1, thetas, i, nh);
        float* tmp = prev0; prev0 = prev1; prev1 = tmp;  // prev1<-prev0, prev0<-t_i
    }

    // 4) y = poly@W2 + b2 -> ypad[N,48]; then log_softmax -> d_out[N,40]
    fc2_wmma_kernel<<<rowTiles / 2, 64, 0, stream>>>(poly, W2, b2, ypad);
    log_softmax40_kernel<<<(N + 255) / 256, 256, 0, stream>>>(ypad, (float*)d_out, N);
}